// DigitCaps2_2010044695158
// MI455X (gfx1250) — compile-verified
//
#include <hip/hip_runtime.h>
#include <math.h>

// Problem constants (from reference setup_inputs)
#define BATCH 128
#define IC    1024   // in_caps
#define ID    16     // in_dim
#define NC    16     // num_caps
#define DC    16     // dim_caps
#define COEF  0.25f  // 1/sqrt(16)

typedef __attribute__((ext_vector_type(2))) float v2f;
typedef __attribute__((ext_vector_type(8))) float v8f;

// ---------------------------------------------------------------------------
// One 16x16x16 f32 tile-matmul step via 4 chained V_WMMA_F32_16X16X4_F32.
//   A[b, i] = x[bbase+b, c, i]   (M=b, K=i)
//   B[i, d] = W[n, c, d, i]      (K=i, N=d)      D[b,d] += A*B
// xa/wa already include the +2*hl lane-half shift, so all K-chunk accesses
// are compile-time immediate offsets from the (incrementing) base pointers.
// ---------------------------------------------------------------------------
__device__ __forceinline__ v8f wmma_step(const float* __restrict__ xa,
                                         const float* __restrict__ wa,
                                         v8f acc) {
#pragma unroll
  for (int k0 = 0; k0 < ID; k0 += 4) {
    const float2 av = *(const float2*)(xa + k0);
    const float2 bv = *(const float2*)(wa + k0);
    v2f a, b;
    a.x = av.x; a.y = av.y;
    b.x = bv.x; b.y = bv.y;
    acc = __builtin_amdgcn_wmma_f32_16x16x4_f32(false, a, false, b,
                                                (short)0, acc, false, false);
  }
  return acc;
}

// ---------------------------------------------------------------------------
// Kernel 0: zero the s accumulator region of the workspace
// ---------------------------------------------------------------------------
__global__ void k_zero(float* __restrict__ p, int n) {
  int t = blockIdx.x * blockDim.x + threadIdx.x;
  if (t < n) p[t] = 0.0f;
}

// ---------------------------------------------------------------------------
// Kernel 1: usum[b,n,d] = sum_c W[n,c,:,:] @ x[b,c,:]
// grid = 128 blocks (8 batch-tiles x 16 n), block = 128 threads (4 waves),
// each wave sweeps 256 c (unrolled x4, pointer-increment addressing),
// LDS tree reduction across the 4 waves.
// ---------------------------------------------------------------------------
__global__ __launch_bounds__(128) void k_usum(const float* __restrict__ x,
                                              const float* __restrict__ W,
                                              float* __restrict__ usum) {
  const int blk   = blockIdx.x;
  const int bt    = blk >> 4;          // batch tile 0..7
  const int n     = blk & 15;          // capsule
  const int tid   = threadIdx.x;
  const int wv    = tid >> 5;          // wave 0..3
  const int lane  = tid & 31;
  const int hl    = lane >> 4;
  const int lm    = lane & 15;
  const int bbase = bt * 16;

  __shared__ float red[4 * 256];

  v8f acc = {};
  const int c0 = wv * 256;
  const float* xa = x + ((size_t)(bbase + lm) * IC + c0) * ID + 2 * hl;
  const float* wa = W + (((size_t)n * IC + c0) * DC + lm) * ID + 2 * hl;
  for (int cc = 0; cc < 256; cc += 4) {
#pragma unroll
    for (int u = 0; u < 4; ++u)
      acc = wmma_step(xa + u * ID, wa + u * (DC * ID), acc);
    xa += 4 * ID;        //   64 B * 4 : immediate-offset friendly
    wa += 4 * DC * ID;   // 1024 B * 4
  }

  // C/D layout: VGPR v, lanes 0-15 -> (b=v, d=lane); lanes 16-31 -> (b=v+8)
#pragma unroll
  for (int v = 0; v < 8; ++v) {
    int b = v + 8 * hl;
    red[wv * 256 + b * 16 + lm] = acc[v];
  }
  __syncthreads();

  for (int e = tid; e < 256; e += 128) {
    float s = red[e] + red[256 + e] + red[512 + e] + red[768 + e];
    int b = e >> 4, d = e & 15;
    usum[(((size_t)(bbase + b)) * NC + n) * DC + d] = s;
  }
}

// ---------------------------------------------------------------------------
// Kernel 2: routing. grid = (16 c-slabs, 8 batch-tiles), block = 512 threads
// (16 waves, wave w owns capsule n = w). Per c: recompute u_hat tile via
// WMMA, scores via LDS transpose, softmax across the 16 n-waves (the
// softmax axis), accumulate s in registers, atomically flush at the end.
// ---------------------------------------------------------------------------
__global__ __launch_bounds__(512) void k_route(const float* __restrict__ x,
                                               const float* __restrict__ W,
                                               const float* __restrict__ Bb,
                                               const float* __restrict__ usum,
                                               float* __restrict__ s_acc) {
  const int bt    = blockIdx.y;        // 0..7
  const int slab  = blockIdx.x;        // 0..15
  const int tid   = threadIdx.x;
  const int w     = tid >> 5;          // wave id == capsule n
  const int lane  = tid & 31;
  const int hl    = lane >> 4;
  const int lm    = lane & 15;
  const int bbase = bt * 16;
  const int n     = w;

  __shared__ float U [16 * 256];   // u_hat tile per wave, [b][d]
  __shared__ float US[16 * 256];   // usum per wave,       [b][d]
  __shared__ float SC[16 * 16];    // scores [n][b]  (cross-wave)
  __shared__ float CF[16 * 16];    // coeff  [n][b]  (wave-private)

  // Preload this wave's usum tile into its LDS region (wave-private)
#pragma unroll
  for (int v = 0; v < 8; ++v) {
    int b = v + 8 * hl;
    US[w * 256 + b * 16 + lm] =
        usum[((size_t)(bbase + b) * NC + n) * DC + lm];
  }

  float s_tile[8];
#pragma unroll
  for (int v = 0; v < 8; ++v) s_tile[v] = 0.0f;

  const int cbeg = slab * 64;
  const float* xa = x + ((size_t)(bbase + lm) * IC + cbeg) * ID + 2 * hl;
  const float* wa = W + (((size_t)n * IC + cbeg) * DC + lm) * ID + 2 * hl;
  const float* bp = Bb + (size_t)n * IC + cbeg;

  for (int ci = 0; ci < 64; ++ci) {
    v8f acc = {};
    acc = wmma_step(xa, wa, acc);
    xa += ID;
    wa += DC * ID;

    // stash u_hat tile [b][d] in wave-private LDS for the d-reduction
#pragma unroll
    for (int v = 0; v < 8; ++v) {
      int b = v + 8 * hl;
      U[w * 256 + b * 16 + lm] = acc[v];
    }

    // scores[b] = COEF * <usum[b,n,:], u_hat[b,n,c,:]>  (lanes 0-15, b = lm)
    if (hl == 0) {
      float sc = 0.0f;
#pragma unroll
      for (int d = 0; d < 16; ++d)
        sc += US[w * 256 + lm * 16 + d] * U[w * 256 + lm * 16 + d];
      SC[w * 16 + lm] = COEF * sc;
    }
    __syncthreads();  // all n-waves published their score row

    // softmax over n (the 16 waves) for this (b, c); lanes 0-15, b = lm
    if (hl == 0) {
      float mx = -1e30f;
#pragma unroll
      for (int nn = 0; nn < 16; ++nn) mx = fmaxf(mx, SC[nn * 16 + lm]);
      float sum = 0.0f;
#pragma unroll
      for (int nn = 0; nn < 16; ++nn) sum += __expf(SC[nn * 16 + lm] - mx);
      float val = __expf(SC[w * 16 + lm] - mx) / sum;
      CF[w * 16 + lm] = val + bp[ci];
    }
    __syncthreads();  // protect SC before next iteration overwrites it

    // s[b,n,:] += coeff[b] * u_hat[b,n,c,:]  (accumulate in WMMA C-layout)
#pragma unroll
    for (int v = 0; v < 8; ++v) {
      int b = v + 8 * hl;
      s_tile[v] += CF[w * 16 + b] * acc[v];
    }
  }

  // flush partial s across c-slabs
#pragma unroll
  for (int v = 0; v < 8; ++v) {
    int b = v + 8 * hl;
    atomicAdd(&s_acc[((size_t)(bbase + b) * NC + n) * DC + lm], s_tile[v]);
  }
}

// ---------------------------------------------------------------------------
// Kernel 3: squash.  out = (1 - exp(-||s||)) * s / sqrt(||s||^2 + 1e-8)
// one thread per (b, n), 16-wide vector in registers.
// ---------------------------------------------------------------------------
__global__ __launch_bounds__(256) void k_squash(const float* __restrict__ s_acc,
                                                float* __restrict__ out) {
  int t = blockIdx.x * blockDim.x + threadIdx.x;  // (b*NC + n), 0..2047
  if (t >= BATCH * NC) return;
  const float* sp = s_acc + (size_t)t * DC;
  float v[16];
  float sq = 0.0f;
#pragma unroll
  for (int d = 0; d < 16; ++d) { v[d] = sp[d]; sq += v[d] * v[d]; }
  float norm  = sqrtf(sq);
  float scale = (1.0f - expf(-norm)) * __frsqrt_rn(sq + 1e-8f);
  float* op = out + (size_t)t * DC;
#pragma unroll
  for (int d = 0; d < 16; ++d) op[d] = v[d] * scale;
}

// ---------------------------------------------------------------------------
extern "C" void kernel_launch(void* const* d_in, const int* in_sizes, int n_in,
                              void* d_out, int out_size, void* d_ws, size_t ws_size,
                              hipStream_t stream) {
  const float* x  = (const float*)d_in[0];   // [128,1024,16]
  const float* W  = (const float*)d_in[1];   // [1,16,1024,16,16]
  const float* Bb = (const float*)d_in[2];   // [16,1,1024]
  float* out = (float*)d_out;                // [128,16,16]

  float* ws    = (float*)d_ws;
  float* usum  = ws;                         // 128*16*16 = 32768 floats
  float* s_acc = ws + 32768;                 // 128*16*16 = 32768 floats

  k_zero  <<<(32768 + 255) / 256, 256, 0, stream>>>(s_acc, 32768);
  k_usum  <<<128, 128, 0, stream>>>(x, W, usum);
  k_route <<<dim3(16, 8), 512, 0, stream>>>(x, W, Bb, usum, s_acc);
  k_squash<<<(BATCH * NC + 255) / 256, 256, 0, stream>>>(s_acc, out);
}